// SMPL_23313082483644
// MI455X (gfx1250) — compile-verified
//
#include <hip/hip_runtime.h>
#include <hip/hip_bf16.h>

typedef __attribute__((ext_vector_type(2))) float v2f;
typedef __attribute__((ext_vector_type(8))) float v8f;

#define V_    6890
#define NJ_   24
#define NB_   10
#define NBP_  12            /* beta padded K (10 -> 12 = 3*4) */
#define NP_   207
#define NJR_  19
#define B_    512
#define N3_   (V_ * 3)      /* 20670 */
#define PFS_  208           /* pose_feature padded stride (207 -> 208 = 52*4) */

__constant__ int c_parents[NJ_] = {0, 0, 0, 0, 1, 2, 3, 4, 5, 6, 7, 8, 9, 9, 9,
                                   12, 13, 14, 16, 17, 18, 19, 20, 21};

// ---------------------------------------------------------------------------
// beta (512x10) -> zero-padded (512x12) so the GEMM K-loop needs no guards
// ---------------------------------------------------------------------------
__global__ __launch_bounds__(256)
void pad_beta_kernel(const float* __restrict__ beta, float* __restrict__ betaP)
{
    const int i = blockIdx.x * 256 + threadIdx.x;
    if (i < B_ * NBP_) {
        const int b = i / NBP_, k = i - b * NBP_;
        betaP[i] = (k < NB_) ? beta[b * NB_ + k] : 0.0f;
    }
}

// ---------------------------------------------------------------------------
// fp32 WMMA GEMM:  C[M,N] = A[M,K] @ B[K,N] + addend[m*addStride + n]
// K must be a multiple of 4 (zero-padded in memory). Each wave owns TWO
// adjacent 16x16 tiles (32 columns), reusing one A fragment per two WMMAs.
// Tail N-tile: B column addresses are CLAMPED (columns are independent in
// WMMA, garbage lands only in never-stored columns); only stores are guarded.
// V_WMMA_F32_16X16X4_F32 VGPR layouts (ISA 7.12.2):
//   A 16x4 : lane L -> row M=L&15, VGPR0=K=(L>>4)*2, VGPR1=K+1
//   B 4x16 : lane L -> col N=L&15, VGPR0=K=(L>>4)*2, VGPR1=K+1
//   D 16x16: VGPR r, lane L -> M=(L>>4)*8+r, N=L&15
// ---------------------------------------------------------------------------
__global__ __launch_bounds__(128)
void gemm_wmma_f32(const float* __restrict__ A, int lda,
                   const float* __restrict__ Bm, int ldb,
                   const float* __restrict__ addend, int addStride,
                   float* __restrict__ Cm,
                   int N, int K)
{
    const int lane = threadIdx.x & 31;
    const int wave = threadIdx.x >> 5;
    const int m0   = blockIdx.y * 16;
    const int n0   = blockIdx.x * 128 + wave * 32;   // 2 tiles per wave

    const int mrow = lane & 15;          // A row for this lane
    const int klo  = (lane >> 4) * 2;    // K sub-offset for this lane
    const int nc0  = n0 + (lane & 15);   // first tile column
    const int nc1  = nc0 + 16;           // second tile column
    const int ncl0 = (nc0 < N) ? nc0 : (N - 1);   // clamped load columns
    const int ncl1 = (nc1 < N) ? nc1 : (N - 1);

    const float* Arow = A + (size_t)(m0 + mrow) * lda + klo;
    const float* Brow = Bm + (size_t)klo * ldb;

    v8f acc0 = {}, acc1 = {};
    for (int kk = 0; kk < K; kk += 4) {
        const v2f a = *(const v2f*)(Arow + kk);          // 8B-aligned b64 load
        const float* B0 = Brow + (size_t)kk * ldb;       // row klo   of k-step
        const float* B1 = B0 + ldb;                      // row klo+1 of k-step
        v2f b0, b1;
        b0.x = B0[ncl0]; b0.y = B1[ncl0];
        b1.x = B0[ncl1]; b1.y = B1[ncl1];
        acc0 = __builtin_amdgcn_wmma_f32_16x16x4_f32(
            false, a, false, b0, (short)0, acc0, false, false);
        acc1 = __builtin_amdgcn_wmma_f32_16x16x4_f32(
            false, a, false, b1, (short)0, acc1, false, false);
    }

    const int mbase = m0 + (lane >> 4) * 8;
    #pragma unroll
    for (int r = 0; r < 8; ++r) {
        const int m = mbase + r;
        float* Crow = Cm + (size_t)m * N;
        const float* Add = addend + (size_t)m * addStride;
        if (nc0 < N) Crow[nc0] = acc0[r] + Add[nc0];
        if (nc1 < N) Crow[nc1] = acc1[r] + Add[nc1];
    }
}

// ---------------------------------------------------------------------------
// J[b,j,c] = sum_v J_regressor[v,j] * v_shaped[b,v,c]   (block per batch)
// ---------------------------------------------------------------------------
__global__ __launch_bounds__(256)
void jreg_kernel(const float* __restrict__ vsh,
                 const float* __restrict__ Jreg,
                 float* __restrict__ Jout)
{
    const int b   = blockIdx.x;
    const int tid = threadIdx.x;
    const int wid = tid >> 5, lane = tid & 31;

    float acc[NJ_ * 3] = {};
    for (int v = tid; v < V_; v += 256) {
        const float vx = vsh[(size_t)b * N3_ + v * 3 + 0];
        const float vy = vsh[(size_t)b * N3_ + v * 3 + 1];
        const float vz = vsh[(size_t)b * N3_ + v * 3 + 2];
        #pragma unroll
        for (int j = 0; j < NJ_; ++j) {
            const float w = Jreg[(size_t)v * NJ_ + j];
            acc[j * 3 + 0] += w * vx;
            acc[j * 3 + 1] += w * vy;
            acc[j * 3 + 2] += w * vz;
        }
    }

    __shared__ float red[8][NJ_ * 3];
    #pragma unroll
    for (int i = 0; i < NJ_ * 3; ++i) {
        float r = acc[i];
        for (int off = 16; off > 0; off >>= 1) r += __shfl_down(r, off, 32);
        if (lane == 0) red[wid][i] = r;
    }
    __syncthreads();
    if (tid < NJ_ * 3) {
        float s = 0.0f;
        #pragma unroll
        for (int w = 0; w < 8; ++w) s += red[w][tid];
        Jout[(size_t)b * (NJ_ * 3) + tid] = s;
    }
}

// ---------------------------------------------------------------------------
// Rodrigues + kinematic chain. One thread per batch.
// Emits pose_feature (stride PFS_, zero-padded) and 3x4 relative xforms A12.
// ---------------------------------------------------------------------------
__global__ __launch_bounds__(256)
void chain_kernel(const float* __restrict__ theta,
                  const float* __restrict__ Jin,
                  float* __restrict__ pf,
                  float* __restrict__ A12)
{
    const int b = blockIdx.x * blockDim.x + threadIdx.x;
    if (b >= B_) return;

    float R[NJ_][9];
    #pragma unroll 1
    for (int j = 0; j < NJ_; ++j) {
        const float tx = theta[b * (NJ_ * 3) + j * 3 + 0];
        const float ty = theta[b * (NJ_ * 3) + j * 3 + 1];
        const float tz = theta[b * (NJ_ * 3) + j * 3 + 2];
        const float ax = tx + 1e-8f, ay = ty + 1e-8f, az = tz + 1e-8f;
        const float ang  = sqrtf(ax * ax + ay * ay + az * az);
        const float half = 0.5f * ang;
        const float s = sinf(half), co = cosf(half);
        float qw = co;
        float qx = s * tx / ang, qy = s * ty / ang, qz = s * tz / ang;
        const float qn = rsqrtf(qw * qw + qx * qx + qy * qy + qz * qz);
        qw *= qn; qx *= qn; qy *= qn; qz *= qn;
        R[j][0] = 1.0f - 2.0f * (qy * qy + qz * qz);
        R[j][1] = 2.0f * (qx * qy - qw * qz);
        R[j][2] = 2.0f * (qx * qz + qw * qy);
        R[j][3] = 2.0f * (qx * qy + qw * qz);
        R[j][4] = 1.0f - 2.0f * (qx * qx + qz * qz);
        R[j][5] = 2.0f * (qy * qz - qw * qx);
        R[j][6] = 2.0f * (qx * qz - qw * qy);
        R[j][7] = 2.0f * (qy * qz + qw * qx);
        R[j][8] = 1.0f - 2.0f * (qx * qx + qy * qy);
    }

    // pose_feature = (R[1:] - I) flattened, padded to stride PFS_
    #pragma unroll 1
    for (int j = 1; j < NJ_; ++j)
        for (int e = 0; e < 9; ++e)
            pf[(size_t)b * PFS_ + (j - 1) * 9 + e] =
                R[j][e] - ((e == 0 || e == 4 || e == 8) ? 1.0f : 0.0f);
    pf[(size_t)b * PFS_ + NP_] = 0.0f;   // K-pad, keeps GEMM guard-free

    float Jt[NJ_][3];
    for (int j = 0; j < NJ_; ++j)
        for (int c = 0; c < 3; ++c)
            Jt[j][c] = Jin[(size_t)b * (NJ_ * 3) + j * 3 + c];

    // Global transforms: G[j] = rotation (0..8, row-major) + translation (9..11)
    float G[NJ_][12];
    // root: R0 @ diag(1,-1,-1) -> negate columns 1 and 2
    G[0][0] = R[0][0]; G[0][1] = -R[0][1]; G[0][2] = -R[0][2];
    G[0][3] = R[0][3]; G[0][4] = -R[0][4]; G[0][5] = -R[0][5];
    G[0][6] = R[0][6]; G[0][7] = -R[0][7]; G[0][8] = -R[0][8];
    G[0][9] = Jt[0][0]; G[0][10] = Jt[0][1]; G[0][11] = Jt[0][2];

    #pragma unroll 1
    for (int i = 1; i < NJ_; ++i) {
        const int p = c_parents[i];
        const float tl0 = Jt[i][0] - Jt[p][0];
        const float tl1 = Jt[i][1] - Jt[p][1];
        const float tl2 = Jt[i][2] - Jt[p][2];
        for (int r = 0; r < 3; ++r) {
            for (int c = 0; c < 3; ++c)
                G[i][r * 3 + c] = G[p][r * 3 + 0] * R[i][0 * 3 + c] +
                                  G[p][r * 3 + 1] * R[i][1 * 3 + c] +
                                  G[p][r * 3 + 2] * R[i][2 * 3 + c];
            G[i][9 + r] = G[p][r * 3 + 0] * tl0 + G[p][r * 3 + 1] * tl1 +
                          G[p][r * 3 + 2] * tl2 + G[p][9 + r];
        }
    }

    // A = [G.R | G.t - G.R @ J_j]   (3x4 row-major, 12 floats per joint)
    #pragma unroll 1
    for (int j = 0; j < NJ_; ++j) {
        for (int r = 0; r < 3; ++r) {
            const float rj = G[j][r * 3 + 0] * Jt[j][0] +
                             G[j][r * 3 + 1] * Jt[j][1] +
                             G[j][r * 3 + 2] * Jt[j][2];
            A12[(size_t)b * (NJ_ * 12) + j * 12 + r * 4 + 0] = G[j][r * 3 + 0];
            A12[(size_t)b * (NJ_ * 12) + j * 12 + r * 4 + 1] = G[j][r * 3 + 1];
            A12[(size_t)b * (NJ_ * 12) + j * 12 + r * 4 + 2] = G[j][r * 3 + 2];
            A12[(size_t)b * (NJ_ * 12) + j * 12 + r * 4 + 3] = G[j][9 + r] - rj;
        }
    }
}

// ---------------------------------------------------------------------------
// Fused skinning + joint regression (block per batch, verts never stored):
//   T = sum_j weights[v,j] * A12[b,j]  (3x4)
//   vert = T @ [v_posed, 1]
//   joints[b,jr,c] = sum_v joint_regressor[v,jr] * vert[c]
// ---------------------------------------------------------------------------
__global__ __launch_bounds__(256)
void skin_kernel(const float* __restrict__ vposed,
                 const float* __restrict__ weights,
                 const float* __restrict__ jointreg,
                 const float* __restrict__ A12,
                 float* __restrict__ out)
{
    const int b   = blockIdx.x;
    const int tid = threadIdx.x;
    const int wid = tid >> 5, lane = tid & 31;

    __shared__ float As[NJ_ * 12];
    for (int i = tid; i < NJ_ * 12; i += 256) As[i] = A12[(size_t)b * (NJ_ * 12) + i];
    __syncthreads();

    float jac[NJR_ * 3] = {};
    for (int v = tid; v < V_; v += 256) {
        float T[12] = {};
        #pragma unroll
        for (int j = 0; j < NJ_; ++j) {
            const float w = weights[(size_t)v * NJ_ + j];
            #pragma unroll
            for (int e = 0; e < 12; ++e) T[e] += w * As[j * 12 + e];
        }
        const float px = vposed[(size_t)b * N3_ + v * 3 + 0];
        const float py = vposed[(size_t)b * N3_ + v * 3 + 1];
        const float pz = vposed[(size_t)b * N3_ + v * 3 + 2];
        const float vx = T[0] * px + T[1] * py + T[2]  * pz + T[3];
        const float vy = T[4] * px + T[5] * py + T[6]  * pz + T[7];
        const float vz = T[8] * px + T[9] * py + T[10] * pz + T[11];
        #pragma unroll
        for (int jr = 0; jr < NJR_; ++jr) {
            const float g = jointreg[(size_t)v * NJR_ + jr];
            jac[jr * 3 + 0] += g * vx;
            jac[jr * 3 + 1] += g * vy;
            jac[jr * 3 + 2] += g * vz;
        }
    }

    __shared__ float red[8][NJR_ * 3];
    #pragma unroll
    for (int i = 0; i < NJR_ * 3; ++i) {
        float r = jac[i];
        for (int off = 16; off > 0; off >>= 1) r += __shfl_down(r, off, 32);
        if (lane == 0) red[wid][i] = r;
    }
    __syncthreads();
    if (tid < NJR_ * 3) {
        float s = 0.0f;
        #pragma unroll
        for (int w = 0; w < 8; ++w) s += red[w][tid];
        out[(size_t)b * (NJR_ * 3) + tid] = s;
    }
}

// ---------------------------------------------------------------------------
extern "C" void kernel_launch(void* const* d_in, const int* in_sizes, int n_in,
                              void* d_out, int out_size, void* d_ws, size_t ws_size,
                              hipStream_t stream)
{
    const float* beta       = (const float*)d_in[0];
    const float* theta      = (const float*)d_in[1];
    const float* v_template = (const float*)d_in[2];
    const float* shapedirs  = (const float*)d_in[3];
    const float* posedirs   = (const float*)d_in[4];
    const float* J_reg      = (const float*)d_in[5];
    const float* joint_reg  = (const float*)d_in[6];
    const float* weights    = (const float*)d_in[7];
    float* out = (float*)d_out;

    float* ws    = (float*)d_ws;
    float* vsh   = ws;                                  // B_*N3_ (v_shaped, then v_posed in place)
    float* Jbuf  = vsh   + (size_t)B_ * N3_;            // B_*NJ_*3
    float* pf    = Jbuf  + (size_t)B_ * NJ_ * 3;        // B_*PFS_
    float* A12   = pf    + (size_t)B_ * PFS_;           // B_*NJ_*12
    float* betaP = A12   + (size_t)B_ * NJ_ * 12;       // B_*NBP_

    // 128 columns per block (4 waves x 2 tiles), 16 rows per block
    dim3 gemm_grid((N3_ + 127) / 128, B_ / 16);         // (162, 32)

    // 0) zero-pad beta K: 10 -> 12
    pad_beta_kernel<<<(B_ * NBP_ + 255) / 256, 256, 0, stream>>>(beta, betaP);
    // 1) v_shaped = beta @ shapedirs + v_template (addend broadcast over rows)
    gemm_wmma_f32<<<gemm_grid, 128, 0, stream>>>(betaP, NBP_, shapedirs, N3_,
                                                 v_template, 0, vsh, N3_, NBP_);
    // 2) J = J_regressor^T . v_shaped
    jreg_kernel<<<B_, 256, 0, stream>>>(vsh, J_reg, Jbuf);
    // 3) Rodrigues + kinematic chain -> pose_feature, A
    chain_kernel<<<2, 256, 0, stream>>>(theta, Jbuf, pf, A12);
    // 4) v_posed = pose_feature @ posedirs + v_shaped (in place over vsh)
    gemm_wmma_f32<<<gemm_grid, 128, 0, stream>>>(pf, PFS_, posedirs, N3_,
                                                 vsh, N3_, vsh, N3_, PFS_);
    // 5) fused skinning + joint regression -> out
    skin_kernel<<<B_, 256, 0, stream>>>(vsh, weights, joint_reg, A12, out);
}